// AnnularDilatedKNN_45612552683642
// MI455X (gfx1250) — compile-verified
//
#include <hip/hip_runtime.h>

// CDNA5 / gfx1250: wave32, WMMA f32 16x16x4 (A 16x4 = 2 VGPRs, B 4x16 = 2 VGPRs, C 16x16 = 8 VGPRs)
typedef __attribute__((ext_vector_type(2))) float v2f;
typedef __attribute__((ext_vector_type(8))) float v8f;
typedef __attribute__((ext_vector_type(4))) unsigned int v4u;
typedef __attribute__((ext_vector_type(4))) int v4i;
typedef __attribute__((ext_vector_type(8))) int v8i;

#define N_PTS 4096
#define TILE 16
#define NTILES (N_PTS / TILE)        // 256 column tiles
#define SPLIT 4                      // column-range segments (1 wave each)
#define SEG_TILES (NTILES / SPLIT)   // 64 tiles per segment wave
#define BLOCK_THREADS (SPLIT * 32)   // 128: 4 waves share one 16-row tile
#define FEAT_C 64
#define KSEL 16                      // dilated ids per point
#define MAXHIT 31                    // need first 31 in-ball hits per row
#define R2 256.0f                    // radius^2 = 16^2

#define HAVE_TDM (__has_builtin(__builtin_amdgcn_tensor_load_to_lds) && \
                  __has_builtin(__builtin_amdgcn_s_wait_tensorcnt))

__launch_bounds__(BLOCK_THREADS)
__global__ void annular_dilated_kernel(const float* __restrict__ xyz,
                                       const float* __restrict__ feat,
                                       float* __restrict__ out) {
  __shared__ float lds_xyz[N_PTS * 3];              // whole batch's xyz: 48 KB
  __shared__ int lds_hits[TILE * SPLIT * 32];       // 31 hit slots / (row, segment): 8 KB
  __shared__ int lds_cnt[TILE * SPLIT];

  const int b    = blockIdx.y;
  const int tid  = threadIdx.x;
  const int seg  = tid >> 5;                        // column segment owned by this wave
  const int lane = tid & 31;
  const int col  = lane & 15;
  const bool hi  = lane >= 16;                      // high half-wave holds K=2,3 of A/B

  const int tile_i = blockIdx.x;                    // row tile 0..255 (16 rows)
  const float* xyz_b  = xyz  + (size_t)b * N_PTS * 3;
  const float* feat_b = feat + (size_t)b * N_PTS * FEAT_C;

  // ================= Stage batch xyz into LDS =================
#if HAVE_TDM
  if (tid < 32) {
    // Tensor DMA: one 1-row tile of 12288 dwords, global -> LDS (TENSORcnt).
    const unsigned long long gaddr = (unsigned long long)(size_t)xyz_b;
    const unsigned lds_off = (unsigned)(size_t)&lds_xyz[0];
    const unsigned n_dw = N_PTS * 3;                // 12288 f32 elements

    v4u g0;
    g0.x = 1u;                                      // count=1, is_restore=0, gather off
    g0.y = lds_off;                                 // lds_addr (bytes)
    g0.z = (unsigned)(gaddr & 0xffffffffu);         // global_addr[31:0]
    g0.w = (unsigned)((gaddr >> 32) & 0x1ffffffu)   // global_addr[56:32]
         | (2u << 30);                              // type = 2 ("image")

    v8i g1 = {
      (int)(2u << 16),                              // wg_mask=0, data_size=2 (4 bytes)
      (int)((n_dw & 0xffffu) << 16),                // tensor_dim0[15:0]
      (int)(((n_dw >> 16) & 0xffffu) | (1u << 16)), // tensor_dim0[31:16], tensor_dim1=1
      (int)(n_dw << 16),                            // tensor_dim1[31:16]=0, tile_dim0
      (int)1,                                       // tile_dim1=1, tile_dim2=0
      (int)n_dw,                                    // tensor_dim0_stride[31:0]
      0, 0                                          // stride[47:32]=0, dim1_stride=0
    };
    v4i g2 = {0, 0, 0, 0};
    v4i g3 = {0, 0, 0, 0};
#if __clang_major__ >= 23
    v8i g4 = {0, 0, 0, 0, 0, 0, 0, 0};
    __builtin_amdgcn_tensor_load_to_lds(g0, g1, g2, g3, g4, 0);
#else
    __builtin_amdgcn_tensor_load_to_lds(g0, g1, g2, g3, 0);
#endif
    __builtin_amdgcn_s_wait_tensorcnt(0);
  }
#else
  for (int t = tid; t < (N_PTS * 3) / 4; t += BLOCK_THREADS)
    ((float4*)lds_xyz)[t] = ((const float4*)xyz_b)[t];
#endif
  __syncthreads();

  // ================= A operand + hoisted row norms (same for all 4 waves) ==========
  const int rp = tile_i * TILE + col;
  const float ax = lds_xyz[rp * 3 + 0];
  const float ay = lds_xyz[rp * 3 + 1];
  const float az = lds_xyz[rp * 3 + 2];
  const float sqA = ax * ax + ay * ay + az * az;
  v2f Amat;
  Amat.x = hi ? az : ax;                            // K=0 (lo) / K=2 (hi)
  Amat.y = hi ? 0.0f : ay;                          // K=1 (lo) / K=3 pad (hi)

  // Loop-invariant row-norm broadcasts, hoisted out of the tile sweep:
  // for C register k, lanes 0-15 need |row k|^2, lanes 16-31 need |row k+8|^2.
  float sAsel[8];
#pragma unroll
  for (int k = 0; k < 8; ++k) {
    const float lo2 = __shfl(sqA, k, 32);
    const float hi2 = __shfl(sqA, k + 8, 32);
    sAsel[k] = hi ? hi2 : lo2;
  }

  const int r = col;                                // row scanned by this lane (dup in hi half)
  const int hitbase = (r * SPLIT + seg) * 32;
  int count = 0;

  // ===== Column-segment sweep: 1 WMMA per 16x16 distance tile, segments partition =====
  const int j_lo = seg * SEG_TILES;
  const int j_hi = j_lo + SEG_TILES;
  for (int j = j_lo; j < j_hi; ++j) {
    const int cp = j * TILE + col;
    const float bx = lds_xyz[cp * 3 + 0];
    const float by = lds_xyz[cp * 3 + 1];
    const float bz = lds_xyz[cp * 3 + 2];
    const float sqB = bx * bx + by * by + bz * bz;
    v2f Bmat;
    Bmat.x = hi ? bz : bx;
    Bmat.y = hi ? 0.0f : by;

    v8f acc = {};
    acc = __builtin_amdgcn_wmma_f32_16x16x4_f32(
        false, Amat, false, Bmat, (short)0, acc, false, false);

    // acc[k]: lanes 0-15 = row k cols 0-15; lanes 16-31 = row k+8 cols 0-15.
    unsigned bal[8];
#pragma unroll
    for (int k = 0; k < 8; ++k) {
      const bool in_ball = (sAsel[k] + sqB - 2.0f * acc[k]) < R2;
      bal[k] = __builtin_amdgcn_ballot_w32(in_ball);
    }

    // Row r's 16-bit mask: bal[r&7], low half if r<8 else high half.
    const unsigned s0 = (r & 1) ? bal[1] : bal[0];
    const unsigned s1 = (r & 1) ? bal[3] : bal[2];
    const unsigned s2 = (r & 1) ? bal[5] : bal[4];
    const unsigned s3 = (r & 1) ? bal[7] : bal[6];
    const unsigned t0 = (r & 2) ? s1 : s0;
    const unsigned t1 = (r & 2) ? s3 : s2;
    const unsigned u  = (r & 4) ? t1 : t0;
    unsigned mask = (r < 8) ? (u & 0xFFFFu) : (u >> 16);

    // Record hits in ascending column order (first MAXHIT of this segment).
    while (mask != 0u && count < MAXHIT) {
      const int bit = __ffs(mask) - 1;
      lds_hits[hitbase + count] = j * TILE + bit;
      ++count;
      mask &= mask - 1u;
    }

    // Wave-uniform early exit once all 16 rows saturated within this segment.
    if (__builtin_amdgcn_ballot_w32(count >= MAXHIT) == 0xFFFFFFFFu)
      break;
  }

  lds_cnt[r * SPLIT + seg] = count;
  __syncthreads();

  // ===== Gather: merge segment hit lists (ascending concat), coalesced stores =====
  float* out_xyz  = out;                                 // [B,3,N,16]
  float* out_feat = out + (size_t)4 * 3 * N_PTS * KSEL;  // [B,64,N,16]

  for (int p = tid; p < TILE * KSEL; p += BLOCK_THREADS) {
    const int row = p >> 4;
    const int k   = p & 15;
    const int n   = tile_i * TILE + row;
    const int slot = (k == 0) ? 0 : (15 + k);           // center, then hits 16..30

    int id = -1, first = -1;
    int rem = slot;
#pragma unroll
    for (int s = 0; s < SPLIT; ++s) {
      const int c = lds_cnt[row * SPLIT + s];
      const int base = (row * SPLIT + s) * 32;
      if (first < 0 && c > 0) first = lds_hits[base];
      if (id < 0) {
        if (rem < c) id = lds_hits[base + rem];
        else rem -= c;
      }
    }
    if (id < 0) id = (first >= 0) ? first : n;          // pad with first hit (self-safe)

    // xyz gather straight out of LDS
#pragma unroll
    for (int c = 0; c < 3; ++c)
      out_xyz[(((size_t)b * 3 + c) * N_PTS + n) * KSEL + k] = lds_xyz[id * 3 + c];

    const float* fp = feat_b + (size_t)id * FEAT_C;
#pragma unroll
    for (int c = 0; c < FEAT_C; ++c)
      out_feat[(((size_t)b * FEAT_C + c) * N_PTS + n) * KSEL + k] = fp[c];
  }
}

extern "C" void kernel_launch(void* const* d_in, const int* in_sizes, int n_in,
                              void* d_out, int out_size, void* d_ws, size_t ws_size,
                              hipStream_t stream) {
  const float* xyz  = (const float*)d_in[0];   // [4,4096,3] f32
  const float* feat = (const float*)d_in[1];   // [4,4096,64] f32
  float* out = (float*)d_out;                  // [4,3,4096,16] ++ [4,64,4096,16] f32

  dim3 grid(NTILES, 4);                        // 1024 blocks x 4 waves = 4096 scan waves
  annular_dilated_kernel<<<grid, BLOCK_THREADS, 0, stream>>>(xyz, feat, out);
}